// AstraMambaWrapper_50989851738657
// MI455X (gfx1250) — compile-verified
//
#include <hip/hip_runtime.h>
#include <math.h>

#define SEQ      16384
#define DMODEL   512
#define DSTATE   16
#define DINNER   1024
#define DTRANK   32
#define NTOTAL   65536
#define NCHUNK   64
#define CHUNK    (SEQ / NCHUNK)   // 256
#define LN_EPS   1e-5f

typedef float v2f __attribute__((ext_vector_type(2)));
typedef float v8f __attribute__((ext_vector_type(8)));

// ---------------------------------------------------------------------------
// fp32 WMMA GEMM: C[SEQ x ND] = A[SEQ x K] @ B[K x ND]  (ldc == ldb == ND)
// One wave computes a (16*MT) x 64 tile: MT A-fragments share 4 B-fragments,
// so each K-step issues 4*MT V_WMMA_F32_16X16X4_F32 with B reused MT times.
// ND/LDA are compile-time so B/C strides fold into immediate offsets.
// A frag (16x4): lane l<16 row m0+l, K=k0+{0,1}; lanes 16..31 K=k0+{2,3}.
// B frag (4x16): vgpr v -> K=k0+v+2*half, lane%16 = col.
// C/D: vgpr r -> row +r (lanes 0-15) / +r+8 (lanes 16-31), lane%16 = col.
// ---------------------------------------------------------------------------
template <int ND, int LDA, int MT, int EPI, bool GATHER>
__global__ __launch_bounds__(256) void wmma_gemm(
    const float* __restrict__ A, const int* __restrict__ gather,
    const float* __restrict__ B, float* __restrict__ C,
    const float* __restrict__ bias, int K)
{
    const int lane = threadIdx.x & 31;
    const int wave = blockIdx.x * (blockDim.x >> 5) + (threadIdx.x >> 5);
    constexpr int tilesN = ND >> 6;
    const int tm = wave / tilesN;
    const int tn = wave % tilesN;
    if (tm >= SEQ / (16 * MT)) return;         // uniform per wave: EXEC stays all-1
    const int m0   = tm * (16 * MT);
    const int n0   = tn << 6;
    const int half = lane >> 4;                // 0 or 1
    const int l16  = lane & 15;

    const float* ap[MT];
#pragma unroll
    for (int m = 0; m < MT; ++m) {
        int r = m0 + 16 * m + l16;
        if (GATHER) r = gather[r];
        ap[m] = A + (size_t)r * LDA + 2 * half;
    }
    const float* bp = B + (size_t)(2 * half) * ND + n0 + l16;

    v8f acc[MT][4];
#pragma unroll
    for (int m = 0; m < MT; ++m)
#pragma unroll
        for (int i = 0; i < 4; ++i) acc[m][i] = (v8f){};

    for (int k0 = 0; k0 < K; k0 += 4) {
        v2f b[4];
#pragma unroll
        for (int i = 0; i < 4; ++i) {          // B rows k0+2*half, k0+2*half+1
            b[i].x = bp[16 * i];
            b[i].y = bp[ND + 16 * i];
        }
#pragma unroll
        for (int m = 0; m < MT; ++m) {
            v2f a;
            a.x = ap[m][0];
            a.y = ap[m][1];
#pragma unroll
            for (int i = 0; i < 4; ++i)
                acc[m][i] = __builtin_amdgcn_wmma_f32_16x16x4_f32(
                    false, a, false, b[i], (short)0, acc[m][i], false, false);
            ap[m] += 4;
        }
        bp += 4 * ND;
    }

#pragma unroll
    for (int m = 0; m < MT; ++m) {
#pragma unroll
        for (int i = 0; i < 4; ++i) {
            const int col = n0 + 16 * i + l16;
            float bv = 0.0f;
            if (EPI == 1) bv = bias[col];
#pragma unroll
            for (int r = 0; r < 8; ++r) {
                float v = acc[m][i][r];
                if (EPI == 1) {                 // softplus(v + b)
                    v += bv;
                    v = (v > 20.0f) ? v : log1pf(expf(v));
                }
                C[(size_t)(m0 + 16 * m + r + 8 * half) * ND + col] = v;
            }
        }
    }
}

// ---------------------------------------------------------------------------
// Depthwise causal conv (width 4) + bias + SiLU.  u = xz[:, :DINNER] (ld 2048)
// ---------------------------------------------------------------------------
__global__ void conv_silu_kernel(const float* __restrict__ xz,
                                 const float* __restrict__ conv_w,
                                 const float* __restrict__ conv_b,
                                 float* __restrict__ uc)
{
    const int c = blockIdx.x * blockDim.x + threadIdx.x;   // channel
    const int t = blockIdx.y;                              // time
    float acc = conv_b[c];
#pragma unroll
    for (int k = 0; k < 4; ++k) {
        const int tt = t - 3 + k;
        if (tt >= 0) acc += xz[(size_t)tt * 2048 + c] * conv_w[c * 4 + k];
    }
    const float sig = 1.0f / (1.0f + __expf(-acc));
    uc[(size_t)t * DINNER + c] = acc * sig;                // silu
}

// ---------------------------------------------------------------------------
// Chunked selective scan, pass 1: per-chunk decay product P and local state L
// ---------------------------------------------------------------------------
__global__ void scan_pass1(const float* __restrict__ dt,
                           const float* __restrict__ uc,
                           const float* __restrict__ dbc,
                           const float* __restrict__ A_log,
                           float* __restrict__ Pbuf,
                           float* __restrict__ Lbuf)
{
    const int c = blockIdx.x * blockDim.x + threadIdx.x;   // channel
    const int j = blockIdx.y;                              // chunk
    float Aneg[DSTATE], P[DSTATE], h[DSTATE];
#pragma unroll
    for (int s = 0; s < DSTATE; ++s) {
        Aneg[s] = -expf(A_log[c * DSTATE + s]);
        P[s] = 1.0f;
        h[s] = 0.0f;
    }
    const int t0 = j * CHUNK;
    for (int t = t0; t < t0 + CHUNK; ++t) {
        const float dtc = dt[(size_t)t * DINNER + c];
        const float du  = dtc * uc[(size_t)t * DINNER + c];
        const float* Bt = dbc + (size_t)t * 64 + DTRANK;
#pragma unroll
        for (int s = 0; s < DSTATE; ++s) {
            const float a = __expf(dtc * Aneg[s]);
            h[s] = a * h[s] + du * Bt[s];
            P[s] *= a;
        }
    }
    const size_t base = ((size_t)j * DINNER + c) * DSTATE;
#pragma unroll
    for (int s = 0; s < DSTATE; ++s) {
        Pbuf[base + s] = P[s];
        Lbuf[base + s] = h[s];
    }
}

// pass 2: sequential combine over the 64 chunks, per (channel, state)
__global__ void scan_pass2(const float* __restrict__ P,
                           const float* __restrict__ L,
                           float* __restrict__ Hin)
{
    const int tid = blockIdx.x * blockDim.x + threadIdx.x; // c*16+s, 0..16383
    float h = 0.0f;
    for (int j = 0; j < NCHUNK; ++j) {
        const size_t o = (size_t)j * (DINNER * DSTATE) + tid;
        Hin[o] = h;
        h = P[o] * h + L[o];
    }
}

// pass 3: re-run each chunk from the correct entry state, emit y = h @ C_t
__global__ void scan_pass3(const float* __restrict__ dt,
                           const float* __restrict__ uc,
                           const float* __restrict__ dbc,
                           const float* __restrict__ A_log,
                           const float* __restrict__ Hin,
                           float* __restrict__ ys)
{
    const int c = blockIdx.x * blockDim.x + threadIdx.x;
    const int j = blockIdx.y;
    float Aneg[DSTATE], h[DSTATE];
    const size_t base = ((size_t)j * DINNER + c) * DSTATE;
#pragma unroll
    for (int s = 0; s < DSTATE; ++s) {
        Aneg[s] = -expf(A_log[c * DSTATE + s]);
        h[s] = Hin[base + s];
    }
    const int t0 = j * CHUNK;
    for (int t = t0; t < t0 + CHUNK; ++t) {
        const float dtc = dt[(size_t)t * DINNER + c];
        const float du  = dtc * uc[(size_t)t * DINNER + c];
        const float* Bt = dbc + (size_t)t * 64 + DTRANK;
        const float* Ct = Bt + DSTATE;
        float y = 0.0f;
#pragma unroll
        for (int s = 0; s < DSTATE; ++s) {
            const float a = __expf(dtc * Aneg[s]);
            h[s] = a * h[s] + du * Bt[s];
            y += h[s] * Ct[s];
        }
        ys[(size_t)t * DINNER + c] = y;
    }
}

// y = (ys + uc * D_skip) * silu(z)   (in place into ys)
__global__ void gating_kernel(float* __restrict__ ys,
                              const float* __restrict__ uc,
                              const float* __restrict__ xz,
                              const float* __restrict__ D_skip)
{
    const size_t i = (size_t)blockIdx.x * blockDim.x + threadIdx.x;
    const int    c = (int)(i & (DINNER - 1));
    const size_t t = i >> 10;
    const float  z = xz[t * 2048 + DINNER + c];
    const float sz = z / (1.0f + __expf(-z));
    ys[i] = (ys[i] + uc[i] * D_skip[c]) * sz;
}

// out = x everywhere (scatter kernel then overwrites the S gathered rows)
__global__ void copy_kernel(const float* __restrict__ x, float* __restrict__ out, size_t n)
{
    const size_t i = (size_t)blockIdx.x * blockDim.x + threadIdx.x;
    if (i < n) out[i] = x[i];
}

// LayerNorm over DMODEL, residual add, scatter to out[idx[r]]
__global__ void ln_scatter_kernel(const float* __restrict__ ych,
                                  const float* __restrict__ x,
                                  const int* __restrict__ idx,
                                  const float* __restrict__ ln_w,
                                  const float* __restrict__ ln_b,
                                  float* __restrict__ out)
{
    __shared__ float red[256];
    const int r  = blockIdx.x;
    const int tx = threadIdx.x;
    const float* yr = ych + (size_t)r * DMODEL;
    const float v0 = yr[tx];
    const float v1 = yr[tx + 256];

    red[tx] = v0 + v1;
    __syncthreads();
    for (int st = 128; st > 0; st >>= 1) {
        if (tx < st) red[tx] += red[tx + st];
        __syncthreads();
    }
    const float mu = red[0] * (1.0f / DMODEL);
    __syncthreads();

    const float d0 = v0 - mu, d1 = v1 - mu;
    red[tx] = d0 * d0 + d1 * d1;
    __syncthreads();
    for (int st = 128; st > 0; st >>= 1) {
        if (tx < st) red[tx] += red[tx + st];
        __syncthreads();
    }
    const float rs = rsqrtf(red[0] * (1.0f / DMODEL) + LN_EPS);

    const int row = idx[r];
    const float* xr = x + (size_t)row * DMODEL;
    float* orow = out + (size_t)row * DMODEL;
    orow[tx]       = xr[tx]       + d0 * rs * ln_w[tx]       + ln_b[tx];
    orow[tx + 256] = xr[tx + 256] + d1 * rs * ln_w[tx + 256] + ln_b[tx + 256];
}

// ---------------------------------------------------------------------------
extern "C" void kernel_launch(void* const* d_in, const int* in_sizes, int n_in,
                              void* d_out, int out_size, void* d_ws, size_t ws_size,
                              hipStream_t stream)
{
    const float* x       = (const float*)d_in[0];
    const int*   seq_idx = (const int*)  d_in[1];
    // d_in[2] seq_mask, d_in[3] node_type: unused by the reference
    const float* W_in    = (const float*)d_in[4];    // 512 x 2048
    const float* conv_w  = (const float*)d_in[5];    // 1024 x 4
    const float* conv_b  = (const float*)d_in[6];    // 1024
    const float* W_xproj = (const float*)d_in[7];    // 1024 x 64
    const float* W_dt    = (const float*)d_in[8];    // 32 x 1024
    const float* b_dt    = (const float*)d_in[9];    // 1024
    const float* A_log   = (const float*)d_in[10];   // 1024 x 16
    const float* D_skip  = (const float*)d_in[11];   // 1024
    const float* W_out   = (const float*)d_in[12];   // 1024 x 512
    const float* ln_w    = (const float*)d_in[13];   // 512
    const float* ln_b    = (const float*)d_in[14];   // 512
    float* out = (float*)d_out;

    // ---- workspace carve-up (floats) ----
    float* w = (float*)d_ws;
    size_t o = 0;
    float* xz  = w + o; o += (size_t)SEQ * 2048;            // 128 MB
    float* uc  = w + o; o += (size_t)SEQ * DINNER;          //  64 MB
    float* dt  = w + o; o += (size_t)SEQ * DINNER;          //  64 MB
    float* dbc = w + o; o += (size_t)SEQ * 64;              //   4 MB
    float* ys  = w + o; o += (size_t)SEQ * DINNER;          //  64 MB
    float* ych = w + o; o += (size_t)SEQ * DMODEL;          //  32 MB
    float* Pb  = w + o; o += (size_t)NCHUNK * DINNER * DSTATE;
    float* Lb  = w + o; o += (size_t)NCHUNK * DINNER * DSTATE;
    float* Hin = w + o; o += (size_t)NCHUNK * DINNER * DSTATE;

    // 0) out = x
    {
        const size_t n = (size_t)NTOTAL * DMODEL;
        copy_kernel<<<dim3((unsigned)((n + 255) / 256)), dim3(256), 0, stream>>>(x, out, n);
    }

    // 1) xz = x[idx] @ W_in   (16384x512 @ 512x2048), 64x64 tiles per wave
    {
        const int waves = (SEQ / 64) * (2048 / 64);          // 8192
        wmma_gemm<2048, 512, 4, 0, true><<<dim3(waves / 8), dim3(256), 0, stream>>>(
            x, seq_idx, W_in, xz, nullptr, DMODEL);
    }

    // 2) depthwise conv + SiLU -> uc
    conv_silu_kernel<<<dim3(DINNER / 256, SEQ), dim3(256), 0, stream>>>(xz, conv_w, conv_b, uc);

    // 3) dbc = uc @ W_xproj   (16384x1024 @ 1024x64), 32x64 tiles
    {
        const int waves = (SEQ / 32) * (64 / 64);            // 512
        wmma_gemm<64, 1024, 2, 0, false><<<dim3(waves / 8), dim3(256), 0, stream>>>(
            uc, nullptr, W_xproj, dbc, nullptr, DINNER);
    }

    // 4) dt = softplus(dbc[:, :32] @ W_dt + b_dt)   (16384x32 @ 32x1024), 32x64 tiles
    {
        const int waves = (SEQ / 32) * (DINNER / 64);        // 8192
        wmma_gemm<DINNER, 64, 2, 1, false><<<dim3(waves / 8), dim3(256), 0, stream>>>(
            dbc, nullptr, W_dt, dt, b_dt, DTRANK);
    }

    // 5) chunked selective scan
    scan_pass1<<<dim3(DINNER / 256, NCHUNK), dim3(256), 0, stream>>>(dt, uc, dbc, A_log, Pb, Lb);
    scan_pass2<<<dim3((DINNER * DSTATE) / 256), dim3(256), 0, stream>>>(Pb, Lb, Hin);
    scan_pass3<<<dim3(DINNER / 256, NCHUNK), dim3(256), 0, stream>>>(dt, uc, dbc, A_log, Hin, ys);

    // 6) gating: ys = (ys + uc*D_skip) * silu(z)
    gating_kernel<<<dim3((SEQ * DINNER) / 256), dim3(256), 0, stream>>>(ys, uc, xz, D_skip);

    // 7) ych = ys @ W_out   (16384x1024 @ 1024x512), 64x64 tiles
    {
        const int waves = (SEQ / 64) * (DMODEL / 64);        // 2048
        wmma_gemm<DMODEL, 1024, 4, 0, false><<<dim3(waves / 8), dim3(256), 0, stream>>>(
            ys, nullptr, W_out, ych, nullptr, DINNER);
    }

    // 8) LayerNorm + residual + scatter into out rows idx
    ln_scatter_kernel<<<dim3(SEQ), dim3(256), 0, stream>>>(ych, x, seq_idx, ln_w, ln_b, out);
}